// OHEMCrossEntropyLoss_4526895530248
// MI455X (gfx1250) — compile-verified
//
#include <hip/hip_runtime.h>
#include <hip/hip_bf16.h>
#include <stdint.h>

#define LDSPTR __attribute__((address_space(3)))

// Byte offset of a generic pointer within the workgroup LDS allocation
// (addrspacecast flat->local extracts the LDS offset).
__device__ __forceinline__ unsigned lds_byte_offset(const void* p) {
  return (unsigned)(unsigned long long)(LDSPTR const char*)p;
}

constexpr int   C_DIM = 128;
constexpr int   WPB   = 8;      // waves per 256-thread block
constexpr float LOG2E = 1.4426950408889634f;
constexpr float LN2   = 0.6931471805599453f;

// ---------------------------------------------------------------------------
// Kernel 1: per-row cross-entropy loss, async global->LDS double-buffered.
// One wave32 per row: 32 lanes x 16B = 512B = one row of 128 f32.
// pred is streamed once -> non-temporal hint keeps it from evicting the
// losses array (4 MB) that later selection passes serve from L2.
// ---------------------------------------------------------------------------
__global__ __launch_bounds__(256) void ohem_loss_kernel(
    const float* __restrict__ pred, const int* __restrict__ target,
    float* __restrict__ losses, int N)
{
  __shared__ float sbuf[WPB][2][C_DIM];   // 8 KB: 2 buffers per wave
  const int lane   = threadIdx.x & 31;
  const int w      = threadIdx.x >> 5;
  const int waveId = blockIdx.x * WPB + w;
  const int stride = gridDim.x * WPB;

  unsigned ldsAddr[2];
  ldsAddr[0] = lds_byte_offset(&sbuf[w][0][0]) + (unsigned)lane * 16u;
  ldsAddr[1] = lds_byte_offset(&sbuf[w][1][0]) + (unsigned)lane * 16u;

  if (waveId >= N) return;

  auto issue = [&](int row, int buf) {
    unsigned goff = (unsigned)row * (unsigned)(C_DIM * 4) + (unsigned)lane * 16u;
    // CDNA5 async DMA: copies 16 bytes from global address saddr + vaddr to the LDS address in vdst; tracked by ASYNCcnt.
    asm volatile("global_load_async_to_lds_b128 %0, %1, %2 th:TH_LOAD_NT"
                 :: "v"(ldsAddr[buf]), "v"(goff), "s"(pred) : "memory");
  };

  issue(waveId, 0);
  int b = 0;
  for (int row = waveId; row < N; row += stride) {
    if (row + stride < N) {
      issue(row + stride, b ^ 1);                      // keep 1 DMA in flight
      asm volatile("s_wait_asynccnt 0x1" ::: "memory");
    } else {
      asm volatile("s_wait_asynccnt 0x0" ::: "memory");
    }

    const float4 v = *reinterpret_cast<const float4*>(&sbuf[w][b][lane * 4]);

    float m = fmaxf(fmaxf(v.x, v.y), fmaxf(v.z, v.w));
    #pragma unroll
    for (int off = 16; off >= 1; off >>= 1)
      m = fmaxf(m, __shfl_xor(m, off, 32));

    float s = __builtin_amdgcn_exp2f((v.x - m) * LOG2E)
            + __builtin_amdgcn_exp2f((v.y - m) * LOG2E)
            + __builtin_amdgcn_exp2f((v.z - m) * LOG2E)
            + __builtin_amdgcn_exp2f((v.w - m) * LOG2E);
    #pragma unroll
    for (int off = 16; off >= 1; off >>= 1)
      s += __shfl_xor(s, off, 32);

    const int   t  = target[row];
    const float pt = sbuf[w][b][t];                    // LDS broadcast read
    if (lane == 0) {
      float loss = m + LN2 * __builtin_amdgcn_logf(s) - pt;  // ln(s)=ln2*log2(s)
      losses[row] = fmaxf(loss, 0.0f);  // keep u32 bit order monotone
    }
    b ^= 1;
  }
}

// ---------------------------------------------------------------------------
// Top-k selection: 4-pass radix select on f32 bit patterns (all values >= 0).
// ---------------------------------------------------------------------------
__global__ void ohem_init_kernel(unsigned* hist, unsigned* state, unsigned* cnt, int K) {
  for (int j = threadIdx.x; j < 1024; j += blockDim.x) hist[j] = 0;
  if (threadIdx.x == 0) { state[0] = 0u; state[1] = (unsigned)K; cnt[0] = 0u; }
}

__global__ __launch_bounds__(256) void ohem_hist_kernel(
    const float* __restrict__ losses, unsigned* __restrict__ hist,
    const unsigned* __restrict__ state, int N, int pass)
{
  __shared__ unsigned lh[256];
  for (int j = threadIdx.x; j < 256; j += blockDim.x) lh[j] = 0;
  __syncthreads();
  const unsigned prefix = state[0];
  const int      shift  = 24 - 8 * pass;
  const unsigned pmask  = (pass == 0) ? 0u : (0xFFFFFFFFu << (shift + 8));
  for (int i = blockIdx.x * blockDim.x + threadIdx.x; i < N;
       i += gridDim.x * blockDim.x) {
    unsigned u = __float_as_uint(losses[i]);
    if ((u & pmask) == (prefix & pmask))
      atomicAdd(&lh[(u >> shift) & 0xFFu], 1u);
  }
  __syncthreads();
  for (int j = threadIdx.x; j < 256; j += blockDim.x)
    if (lh[j]) atomicAdd(&hist[j], lh[j]);
}

__global__ void ohem_pick_kernel(const unsigned* __restrict__ hist,
                                 unsigned* __restrict__ state, int pass) {
  unsigned kk = state[1];
  unsigned run = 0;
  int d = 255;
  for (; d > 0; --d) {
    unsigned h = hist[d];
    if (run + h >= kk) break;
    run += h;
  }
  state[0] |= ((unsigned)d) << (24 - 8 * pass);
  state[1]  = kk - run;
}

// ---------------------------------------------------------------------------
// Deterministic sum of all losses strictly above the k-th value.
// ---------------------------------------------------------------------------
__global__ __launch_bounds__(256) void ohem_sum_kernel(
    const float* __restrict__ losses, const unsigned* __restrict__ state,
    float* __restrict__ partials, unsigned* __restrict__ cnt, int N)
{
  __shared__ float    wsum[WPB];
  __shared__ unsigned wcnt[WPB];
  const unsigned thresh = state[0];
  float acc = 0.0f; unsigned c = 0;
  for (int i = blockIdx.x * blockDim.x + threadIdx.x; i < N;
       i += gridDim.x * blockDim.x) {
    float l = losses[i];
    if (__float_as_uint(l) > thresh) { acc += l; c++; }
  }
  #pragma unroll
  for (int off = 16; off >= 1; off >>= 1) {
    acc += __shfl_xor(acc, off, 32);
    c   += __shfl_xor(c,   off, 32);
  }
  const int lane = threadIdx.x & 31, w = threadIdx.x >> 5;
  if (lane == 0) { wsum[w] = acc; wcnt[w] = c; }
  __syncthreads();
  if (threadIdx.x == 0) {
    float bs = 0.0f; unsigned bc = 0;
    for (int j = 0; j < WPB; ++j) { bs += wsum[j]; bc += wcnt[j]; }
    partials[blockIdx.x] = bs;          // fixed slot -> deterministic
    atomicAdd(cnt, bc);                 // integer add -> deterministic
  }
}

__global__ void ohem_final_kernel(const float* __restrict__ partials, int nPartials,
                                  const unsigned* __restrict__ state,
                                  const unsigned* __restrict__ cnt,
                                  float* __restrict__ out, int K)
{
  double s = 0.0;
  for (int i = 0; i < nPartials; ++i) s += (double)partials[i];  // fixed order
  const unsigned cg = cnt[0];
  const float    tv = __uint_as_float(state[0]);       // k-th largest loss
  const double total = s + (double)(K - (int)cg) * (double)tv; // ties at k-th
  out[0] = (float)(total / (double)K);
}

// ---------------------------------------------------------------------------
extern "C" void kernel_launch(void* const* d_in, const int* in_sizes, int n_in,
                              void* d_out, int out_size, void* d_ws, size_t ws_size,
                              hipStream_t stream) {
  (void)n_in; (void)out_size; (void)ws_size;
  const float* pred   = (const float*)d_in[0];
  const int*   target = (const int*)d_in[1];
  const int    N      = in_sizes[1];
  const int    K      = (int)((double)N * 0.7);   // matches python int(N*0.7)

  float*    losses   = (float*)d_ws;              // N floats
  unsigned* hist     = (unsigned*)(losses + N);   // 4 passes x 256
  unsigned* state    = hist + 1024;               // [prefix, kk]
  unsigned* cnt      = state + 2;                 // count strictly greater
  float*    partials = (float*)(cnt + 2);         // SUM_BLOCKS floats

  constexpr int SUM_BLOCKS = 256;                 // short deterministic tail

  ohem_init_kernel<<<1, 256, 0, stream>>>(hist, state, cnt, K);
  ohem_loss_kernel<<<8192, 256, 0, stream>>>(pred, target, losses, N);
  for (int pass = 0; pass < 4; ++pass) {
    ohem_hist_kernel<<<1024, 256, 0, stream>>>(losses, hist + pass * 256, state, N, pass);
    ohem_pick_kernel<<<1, 1, 0, stream>>>(hist + pass * 256, state, pass);
  }
  ohem_sum_kernel<<<SUM_BLOCKS, 256, 0, stream>>>(losses, state, partials, cnt, N);
  ohem_final_kernel<<<1, 1, 0, stream>>>(partials, SUM_BLOCKS, state, cnt, (float*)d_out, K);
}